// Decoder_80590766342167
// MI455X (gfx1250) — compile-verified
//
#include <hip/hip_runtime.h>
#include <hip/hip_bf16.h>

// Problem constants (from reference): S=4096, B=64, H=256
#define S_LEN 4096
#define B_DIM 64
#define H_DIM 256
#define N_ROWS (S_LEN * B_DIM)      // 262144 rows of the GEMV
#define ROWS_PER_WAVE 16            // one 16x16x4 WMMA tile of rows per wave
#define WAVES_PER_BLOCK 8           // 256 threads
#define N_TILES (N_ROWS / ROWS_PER_WAVE)          // 16384
#define N_BLOCKS (N_TILES / WAVES_PER_BLOCK)      // 2048

typedef float v2f __attribute__((ext_vector_type(2)));
typedef float v4f __attribute__((ext_vector_type(4)));
typedef float v8f __attribute__((ext_vector_type(8)));

// GEMV via V_WMMA_F32_16X16X4_F32 (fp32-exact, memory-bound workload).
// Wave w of block computes preds for rows [tile*16, tile*16+16) and the
// squared-error contribution vs outputs^T; loss reduced per-block then one
// global atomic per block.
__global__ void __launch_bounds__(256)
decoder_gemv_wmma(const float* __restrict__ hidden,   // [S,B,H]
                  const float* __restrict__ outputs,  // [B,S]
                  const float* __restrict__ W,        // [H]
                  const float* __restrict__ bias,     // [1]
                  float* __restrict__ out)            // [1 + S*B]
{
    __shared__ float sW[H_DIM];
    __shared__ float sLoss;

    const int tid = threadIdx.x;
    // Preload W (1 KB) into LDS; it feeds the B operand every iteration.
    sW[tid] = W[tid];                 // blockDim == H_DIM == 256
    if (tid == 0) sLoss = 0.0f;
    __syncthreads();

    const int lane = tid & 31;
    const int wave = tid >> 5;
    const int tile = blockIdx.x * WAVES_PER_BLOCK + wave;
    const int r0   = tile * ROWS_PER_WAVE;

    const int m    = lane & 15;       // A-matrix row owned by this lane
    const int half = lane >> 4;       // 0: holds K={0,1}; 1: holds K={2,3}

    // Each lane streams float4s from its row; the 4-float offset split by
    // lane-half makes one B128 load supply the A operands of two K=4 WMMAs
    // over the permuted K-sets {h,h+1,h+4,h+5} and {h+2,h+3,h+6,h+7}.
    const float* rowp = hidden + (size_t)(r0 + m) * H_DIM + half * 4;
    const float* wp   = sW + half * 4;

    v8f c = {};  // 16x16 f32 accumulator: D[j][n] (all n equal) = pred row j

    #pragma unroll 4
    for (int h = 0; h < H_DIM; h += 8) {
        v4f a = __builtin_nontemporal_load((const v4f*)(rowp + h)); // stream once
        v4f w = *(const v4f*)(wp + h);                              // LDS broadcast
        v2f a0 = {a.x, a.y}, a1 = {a.z, a.w};
        v2f b0 = {w.x, w.y}, b1 = {w.z, w.w};
        c = __builtin_amdgcn_wmma_f32_16x16x4_f32(false, a0, false, b0,
                                                  (short)0, c, false, false);
        c = __builtin_amdgcn_wmma_f32_16x16x4_f32(false, a1, false, b1,
                                                  (short)0, c, false, false);
    }

    // D layout: VGPR j holds M=j (lanes 0-15) / M=j+8 (lanes 16-31), N=lane%16.
    // B columns are identical, so lane 0 owns preds r0..r0+7 and lane 16 owns
    // preds r0+8..r0+15, each in 8 consecutive VGPRs.
    const float bv = bias[0];
    if ((lane & 15) == 0) {
        const int rbase = r0 + 8 * half;
        float ls = 0.0f;
        #pragma unroll
        for (int j = 0; j < 8; ++j) {
            const int r = rbase + j;
            const float p = c[j] + bv;
            out[1 + r] = p;                         // preds.reshape(-1), s-major
            const int s = r >> 6;                   // r / B
            const int b = r & (B_DIM - 1);          // r % B
            const float d = p - outputs[(size_t)b * S_LEN + s];
            ls += d * d;
        }
        atomicAdd(&sLoss, ls);
    }
    __syncthreads();
    if (tid == 0) atomicAdd(out, sLoss);
}

extern "C" void kernel_launch(void* const* d_in, const int* in_sizes, int n_in,
                              void* d_out, int out_size, void* d_ws, size_t ws_size,
                              hipStream_t stream) {
    // setup_inputs() order: outputs [B,S], hidden [S,B,H], W [1,H], b [1]
    const float* outputs = (const float*)d_in[0];
    const float* hidden  = (const float*)d_in[1];
    const float* W       = (const float*)d_in[2];
    const float* bias    = (const float*)d_in[3];
    float* out = (float*)d_out;

    // out[0] = loss accumulator: must start at zero every call (graph-safe memset node).
    hipMemsetAsync(d_out, 0, sizeof(float), stream);

    decoder_gemv_wmma<<<N_BLOCKS, 256, 0, stream>>>(hidden, outputs, W, bias, out);
}